// Attention_20091857010765
// MI455X (gfx1250) — compile-verified
//
#include <hip/hip_runtime.h>

#define BATCH 8
#define CDIM  128
#define HF    128
#define WF    128
#define SFULL (HF*WF)   // 16384
#define HH    64
#define WH    64
#define SHALF (HH*WH)   // 4096
#define NHEADS 8
#define CH    16        // CDIM / NHEADS
#define CA    8         // (CDIM/2) / NHEADS

typedef __attribute__((ext_vector_type(2))) float v2f;
typedef __attribute__((ext_vector_type(8))) float v8f;

__device__ __forceinline__ v8f wmma4(v2f a, v2f b, v8f c) {
    // D = A(16x4) * B(4x16) + C, fp32 throughout
    return __builtin_amdgcn_wmma_f32_16x16x4_f32(
        false, a, false, b, (short)0, c, false, false);
}

// ---------------------------------------------------------------------------
// Kernel 0: transpose wk [co][ci][tap] -> wkT [tap][co][ci]  (147K floats)
// makes conv-k A-fragments contiguous in K
// ---------------------------------------------------------------------------
__global__ void transpose_wk(const float* __restrict__ wk,
                             float* __restrict__ wkT) {
    int idx = blockIdx.x * blockDim.x + threadIdx.x;   // over CDIM*CDIM*9
    if (idx >= CDIM * CDIM * 9) return;
    int tap = idx % 9;
    int rest = idx / 9;
    int ci = rest & (CDIM - 1);
    int co = rest >> 7;
    wkT[((size_t)tap * CDIM + co) * CDIM + ci] = wk[idx];
}

// ---------------------------------------------------------------------------
// Kernel 1: fused depthwise 3x3 stride-2 convs for q and a_dw (shared x reads)
// ---------------------------------------------------------------------------
__global__ void dwconv_s2(const float* __restrict__ x,
                          const float* __restrict__ wq,
                          const float* __restrict__ wa,
                          float* __restrict__ outq,
                          float* __restrict__ outa) {
    int idx = blockIdx.x * blockDim.x + threadIdx.x;   // over B*C*SHALF
    if (idx >= BATCH * CDIM * SHALF) return;
    int p  = idx & (SHALF - 1);
    int bc = idx >> 12;                 // SHALF == 4096
    int c  = bc & (CDIM - 1);
    int y  = p >> 6, xo = p & 63;
    const float* xin = x + (size_t)bc * SFULL;
    float accq = 0.f, acca = 0.f;
#pragma unroll
    for (int ky = 0; ky < 3; ++ky) {
        int iy = 2 * y + ky - 1;
        if ((unsigned)iy >= (unsigned)HF) continue;
#pragma unroll
        for (int kx = 0; kx < 3; ++kx) {
            int ix = 2 * xo + kx - 1;
            if ((unsigned)ix >= (unsigned)WF) continue;
            float xv = xin[iy * WF + ix];
            accq += xv * wq[c * 9 + ky * 3 + kx];
            acca += xv * wa[c * 9 + ky * 3 + kx];
        }
    }
    outq[idx] = accq;
    outa[idx] = acca;
}

// ---------------------------------------------------------------------------
// Kernel 2: k = conv3x3/s2 dense as implicit GEMM, tap-major K ordering.
// iy is wave-uniform (scalar tap skip, EXEC stays all-1s for WMMA);
// ix boundary handled with branchless cndmask on a clamped address.
// ---------------------------------------------------------------------------
__global__ void convk_wmma(const float* __restrict__ x,
                           const float* __restrict__ wkT,  // [9][co][ci]
                           float* __restrict__ kout) {
    int wave  = threadIdx.x >> 5;
    int lane  = threadIdx.x & 31;
    int ntile = blockIdx.x * 4 + wave;   // 0..255  (SHALF/16)
    int mtile = blockIdx.y;              // 0..7    (CDIM/16)
    int b     = blockIdx.z;
    int nm    = lane & 15;
    int khalf = (lane >> 4) << 1;        // 0 or 2
    int y  = ntile >> 2;                 // 4 tiles per 64-px output row
    int xo = ((ntile & 3) << 4) + nm;
    int co = mtile * 16 + nm;
    const float* xin = x + (size_t)b * CDIM * SFULL;
    v8f acc = {};
#pragma unroll
    for (int ky = 0; ky < 3; ++ky) {
        int iy = 2 * y + ky - 1;                         // wave-uniform
        if ((unsigned)iy >= (unsigned)HF) continue;      // scalar branch
#pragma unroll
        for (int kx = 0; kx < 3; ++kx) {
            int tap = ky * 3 + kx;
            int ix = 2 * xo + kx - 1;
            bool bvalid = (unsigned)ix < (unsigned)WF;   // per-lane, branchless
            int ixc = bvalid ? ix : 0;
            const float* bbase = xin + (size_t)iy * WF + ixc;
            const float* abase = wkT + ((size_t)tap * CDIM + co) * CDIM;
#pragma unroll 8
            for (int ci = 0; ci < CDIM; ci += 4) {
                int ka = ci + khalf;
                v2f af = *(const v2f*)(abase + ka);      // contiguous b64
                float b0 = bbase[(size_t)ka * SFULL];
                float b1 = bbase[(size_t)(ka + 1) * SFULL];
                v2f bf;
                bf.x = bvalid ? b0 : 0.f;
                bf.y = bvalid ? b1 : 0.f;
                acc = wmma4(af, bf, acc);
            }
        }
    }
    float* outp = kout + ((size_t)b * CDIM + mtile * 16) * SHALF + ntile * 16;
    int rbase = (lane >> 4) * 8;
#pragma unroll
    for (int i = 0; i < 8; ++i)
        outp[(size_t)(rbase + i) * SHALF + (lane & 15)] = acc[i];
}

// ---------------------------------------------------------------------------
// Kernel 3: WMMA GEMM  out[b,M,N] = A[M,K] * Bm[b,K,N]  (* mul[b,M,N])
// K compile-time -> fully unrolled WMMA stream; NT -> nontemporal stores
// ---------------------------------------------------------------------------
template <int K, bool NT>
__global__ void gemm_wmma(const float* __restrict__ A,
                          const float* __restrict__ Bm,
                          const float* __restrict__ mul,
                          float* __restrict__ Cm,
                          int M, int N) {
    int wave  = threadIdx.x >> 5;
    int lane  = threadIdx.x & 31;
    int ntile = blockIdx.x * 4 + wave;
    int mtile = blockIdx.y;
    int b     = blockIdx.z;
    int col   = lane & 15;
    int khalf = (lane >> 4) << 1;
    int mrow  = mtile * 16 + (lane & 15);
    const float* Ap = A + (size_t)mrow * K;
    const float* Bp = Bm + (size_t)b * K * N + ntile * 16 + col;
    v8f acc = {};
#pragma unroll
    for (int k0 = 0; k0 < K; k0 += 4) {
        int ka = k0 + khalf;
        v2f af = *(const v2f*)(Ap + ka);                 // contiguous b64
        v2f bf;
        bf.x = Bp[(size_t)ka * N];
        bf.y = Bp[(size_t)(ka + 1) * N];
        acc = wmma4(af, bf, acc);
    }
    size_t outbase = (size_t)b * M * N + (size_t)(mtile * 16) * N + ntile * 16;
    int rbase = (lane >> 4) * 8;
#pragma unroll
    for (int i = 0; i < 8; ++i) {
        size_t oi = outbase + (size_t)(rbase + i) * N + col;
        float val = acc[i];
        if (mul) val *= mul[oi];
        if (NT) __builtin_nontemporal_store(val, &Cm[oi]);
        else    Cm[oi] = val;
    }
}

// ---------------------------------------------------------------------------
// Kernel 4: row-wise L2 normalization (row length = ncols), in place
// ---------------------------------------------------------------------------
__global__ void l2norm_rows(float* __restrict__ data, int ncols) {
    __shared__ float red[256];
    int row = blockIdx.x;
    float* p = data + (size_t)row * ncols;
    float s = 0.f;
    for (int i = threadIdx.x; i < ncols; i += blockDim.x) { float v = p[i]; s += v * v; }
    red[threadIdx.x] = s;
    __syncthreads();
    for (int off = 128; off > 0; off >>= 1) {
        if ((int)threadIdx.x < off) red[threadIdx.x] += red[threadIdx.x + off];
        __syncthreads();
    }
    float inv = 1.f / fmaxf(sqrtf(red[0]), 1e-12f);
    for (int i = threadIdx.x; i < ncols; i += blockDim.x) p[i] *= inv;
}

// ---------------------------------------------------------------------------
// Kernel 5: tiny attention per (b,h): qa/ak dots, softmaxes, Mcomb = Aa @ Ak
// ---------------------------------------------------------------------------
__global__ void attn_small(const float* __restrict__ q,   // [B, C, SHALF]
                           const float* __restrict__ k,   // [B, C, SHALF]
                           const float* __restrict__ a,   // [B, C/2, SHALF]
                           const float* __restrict__ ta,  // [NHEADS]
                           const float* __restrict__ tv,  // [NHEADS]
                           float* __restrict__ Mc) {      // [B*NHEADS, 16, 16]
    int bh = blockIdx.x; int b = bh >> 3, h = bh & 7;
    int tid = threadIdx.x;   // 128 threads
    __shared__ float Aattn[16][8];
    __shared__ float Kattn[8][16];
    {   // qa[c][d] = q_row(c) . a_row(d)
        int c = tid >> 3, d = tid & 7;
        const float4* qr = (const float4*)(q + ((size_t)b * CDIM + h * CH + c) * SHALF);
        const float4* ar = (const float4*)(a + ((size_t)b * (CDIM / 2) + h * CA + d) * SHALF);
        float s = 0.f;
        for (int i = 0; i < SHALF / 4; ++i) {
            float4 u = qr[i], w = ar[i];
            s += u.x * w.x + u.y * w.y + u.z * w.z + u.w * w.w;
        }
        Aattn[c][d] = s * ta[h];
    }
    {   // ak[d][c] = a_row(d) . k_row(c)
        int d = tid >> 4, c = tid & 15;
        const float4* ar = (const float4*)(a + ((size_t)b * (CDIM / 2) + h * CA + d) * SHALF);
        const float4* kr = (const float4*)(k + ((size_t)b * CDIM + h * CH + c) * SHALF);
        float s = 0.f;
        for (int i = 0; i < SHALF / 4; ++i) {
            float4 u = ar[i], w = kr[i];
            s += u.x * w.x + u.y * w.y + u.z * w.z + u.w * w.w;
        }
        Kattn[d][c] = s * tv[h];
    }
    __syncthreads();
    if (tid < 16) {                       // softmax over d (last axis of bhcd)
        float mx = -1e30f;
        for (int d = 0; d < 8; ++d) mx = fmaxf(mx, Aattn[tid][d]);
        float sum = 0.f;
        for (int d = 0; d < 8; ++d) { float e = __expf(Aattn[tid][d] - mx); Aattn[tid][d] = e; sum += e; }
        for (int d = 0; d < 8; ++d) Aattn[tid][d] /= sum;
    } else if (tid < 24) {                // softmax over c (last axis of bhdc)
        int d = tid - 16;
        float mx = -1e30f;
        for (int c = 0; c < 16; ++c) mx = fmaxf(mx, Kattn[d][c]);
        float sum = 0.f;
        for (int c = 0; c < 16; ++c) { float e = __expf(Kattn[d][c] - mx); Kattn[d][c] = e; sum += e; }
        for (int c = 0; c < 16; ++c) Kattn[d][c] /= sum;
    }
    __syncthreads();
    // Mcomb[c][c2] = sum_d Aattn[c][d] * Kattn[d][c2]  (collapses both einsums)
    for (int rep = 0; rep < 2; ++rep) {
        int e = tid + rep * 128;
        int c = e >> 4, c2 = e & 15;
        float s = 0.f;
        for (int d = 0; d < 8; ++d) s += Aattn[c][d] * Kattn[d][c2];
        Mc[(size_t)bh * 256 + c * 16 + c2] = s;
    }
}

// ---------------------------------------------------------------------------
// Kernel 6: out[b,h] = Mcomb[16x16] @ v_head[16 x 16384]   (rank-16, WMMA)
// ---------------------------------------------------------------------------
__global__ void apply_attn(const float* __restrict__ Mc,   // [B*NHEADS,16,16]
                           const float* __restrict__ v,    // [B, C, SFULL]
                           float* __restrict__ outbuf) {   // [B, C, SFULL]
    int wave  = threadIdx.x >> 5;
    int lane  = threadIdx.x & 31;
    int ntile = blockIdx.x * 4 + wave;    // 0..1023
    int bh    = blockIdx.y;               // 0..63
    int b = bh >> 3, h = bh & 7;
    int col   = lane & 15;
    int khalf = (lane >> 4) << 1;
    int m     = lane & 15;
    const float* A  = Mc + (size_t)bh * 256 + m * 16;
    const float* Bp = v + ((size_t)b * CDIM + h * CH) * SFULL + ntile * 16 + col;
    v8f acc = {};
#pragma unroll
    for (int k0 = 0; k0 < 16; k0 += 4) {
        int ka = k0 + khalf;
        v2f af = *(const v2f*)(A + ka);
        v2f bf;
        bf.x = Bp[(size_t)ka * SFULL];
        bf.y = Bp[(size_t)(ka + 1) * SFULL];
        acc = wmma4(af, bf, acc);
    }
    float* op = outbuf + ((size_t)b * CDIM + h * CH) * SFULL + ntile * 16;
    int rbase = (lane >> 4) * 8;
#pragma unroll
    for (int i = 0; i < 8; ++i)
        op[(size_t)(rbase + i) * SFULL + col] = acc[i];
}

// ---------------------------------------------------------------------------
extern "C" void kernel_launch(void* const* d_in, const int* in_sizes, int n_in,
                              void* d_out, int out_size, void* d_ws, size_t ws_size,
                              hipStream_t stream) {
    const float* x     = (const float*)d_in[0];
    const float* illu  = (const float*)d_in[1];
    const float* wq    = (const float*)d_in[2];
    const float* wk    = (const float*)d_in[3];
    const float* wa_dw = (const float*)d_in[4];
    const float* wa_pw = (const float*)d_in[5];
    const float* wv    = (const float*)d_in[6];
    const float* wout  = (const float*)d_in[7];
    const float* ta    = (const float*)d_in[8];
    const float* tv    = (const float*)d_in[9];

    float* ws   = (float*)d_ws;
    float* qb   = ws;                                   // [B,C,SHALF]
    float* kb   = qb  + (size_t)BATCH * CDIM * SHALF;   // [B,C,SHALF]
    float* adw  = kb  + (size_t)BATCH * CDIM * SHALF;   // [B,C,SHALF]
    float* ab   = adw + (size_t)BATCH * CDIM * SHALF;   // [B,C/2,SHALF]
    float* vb   = ab  + (size_t)BATCH * (CDIM / 2) * SHALF; // [B,C,SFULL]
    float* obuf = vb  + (size_t)BATCH * CDIM * SFULL;   // [B,C,SFULL]
    float* mc   = obuf + (size_t)BATCH * CDIM * SFULL;  // [B*H,16,16]
    float* wkT  = mc + (size_t)BATCH * NHEADS * 256;    // [9,C,C]

    // 0. weight transpose for implicit-GEMM conv
    transpose_wk<<<(CDIM * CDIM * 9 + 255) / 256, 256, 0, stream>>>(wk, wkT);
    // 1. depthwise q & a_dw
    dwconv_s2<<<(BATCH * CDIM * SHALF) / 256, 256, 0, stream>>>(x, wq, wa_dw, qb, adw);
    // 2. k = conv3x3/s2 dense (implicit GEMM, WMMA)
    convk_wmma<<<dim3(SHALF / 64, CDIM / 16, BATCH), 128, 0, stream>>>(x, wkT, kb);
    // 3. a = pointwise 1x1 (WMMA)
    gemm_wmma<CDIM, false><<<dim3(SHALF / 64, (CDIM / 2) / 16, BATCH), 128, 0, stream>>>(
        wa_pw, adw, nullptr, ab, CDIM / 2, SHALF);
    // 4. v = wv*x  (*) illu   (WMMA + fused elementwise)
    gemm_wmma<CDIM, false><<<dim3(SFULL / 64, CDIM / 16, BATCH), 128, 0, stream>>>(
        wv, x, illu, vb, CDIM, SFULL);
    // 5. L2 norms over s
    l2norm_rows<<<BATCH * CDIM, 256, 0, stream>>>(qb, SHALF);
    l2norm_rows<<<BATCH * CDIM, 256, 0, stream>>>(kb, SHALF);
    l2norm_rows<<<BATCH * (CDIM / 2), 256, 0, stream>>>(ab, SHALF);
    // 6. tiny attention matrices + softmaxes + combine
    attn_small<<<BATCH * NHEADS, 128, 0, stream>>>(qb, kb, ab, ta, tv, mc);
    // 7. rank-16 application to v (WMMA)
    apply_attn<<<dim3(SFULL / 64, BATCH * NHEADS), 128, 0, stream>>>(mc, vb, obuf);
    // 8. final 1x1 conv (WMMA, nontemporal stores) -> d_out
    gemm_wmma<CDIM, true><<<dim3(SFULL / 64, CDIM / 16, BATCH), 128, 0, stream>>>(
        wout, obuf, nullptr, (float*)d_out, CDIM, SFULL);
}